// GraphPerception_7275674599908
// MI455X (gfx1250) — compile-verified
//
#include <hip/hip_runtime.h>
#include <hip/hip_bf16.h>

// ---------------------------------------------------------------------------
// GNN message passing on MI455X (gfx1250, wave32).
// fp32 WMMA (V_WMMA_F32_16X16X4_F32) for all dense GEMMs; fp32 HW atomics for
// the scatter phases. Edge projection is algebraically hoisted:
//   segsum(edge_attr@Wes + bes, col) == segsum(edge_attr, col)@Wes + deg*bes
// W is staged in LDS in pair-interleaved (fragment-ready) layout so each WMMA
// B fragment is a single aligned ds_load_b64. Waves are M-blocked x2 so each
// B fragment feeds two WMMAs.
// ---------------------------------------------------------------------------

#define NNODES 100000
#define NEDGES 600000
#define HID    128
#define EDIM   16
#define NGRAPH 64

typedef float v2f __attribute__((ext_vector_type(2)));
typedef float v8f __attribute__((ext_vector_type(8)));

__device__ __forceinline__ void atomicAddF(float* p, float v) {
    __hip_atomic_fetch_add(p, v, __ATOMIC_RELAXED, __HIP_MEMORY_SCOPE_AGENT);
}

__device__ __forceinline__ v8f wmma_f32_4(v2f a, v2f b, v8f c) {
    return __builtin_amdgcn_wmma_f32_16x16x4_f32(false, a, false, b, (short)0, c,
                                                 false, false);
}

// --------------------------- utility: zero fill ----------------------------
__global__ void gp_zero_kernel(float* __restrict__ p, long n) {
    long i = (long)blockIdx.x * blockDim.x + threadIdx.x;
    long stride = (long)gridDim.x * blockDim.x;
    for (; i < n; i += stride) p[i] = 0.0f;
}

// ---------------- one-time: in-degree + per-node edge_attr sum -------------
__global__ __launch_bounds__(256) void gp_edge_pre_kernel(
    const float* __restrict__ edge_attr, const int* __restrict__ col,
    float* __restrict__ eagg, float* __restrict__ deg, int nEdges) {
    long gid = (long)blockIdx.x * blockDim.x + threadIdx.x;
    int e = (int)(gid >> 4);
    int comp = (int)(gid & 15);
    if (e >= nEdges) return;
    int c = col[e];
    atomicAddF(&eagg[(size_t)c * EDIM + comp], edge_attr[(size_t)e * EDIM + comp]);
    if (comp == 0) atomicAddF(&deg[c], 1.0f);
}

__global__ __launch_bounds__(256) void gp_count_kernel(
    const int* __restrict__ batch, float* __restrict__ cnt, int n) {
    int i = blockIdx.x * blockDim.x + threadIdx.x;
    if (i < n) atomicAddF(&cnt[batch[i]], 1.0f);
}

// -------------------- per-layer: scatter x[row] -> agg[col] ----------------
// one wave per edge: 32 lanes x float4 = 128 floats
__global__ __launch_bounds__(256) void gp_scatter_kernel(
    const float* __restrict__ xsrc, float* __restrict__ agg,
    const int* __restrict__ row, const int* __restrict__ col, int nEdges) {
    int wid = (int)(((long)blockIdx.x * blockDim.x + threadIdx.x) >> 5);
    int lane = threadIdx.x & 31;
    if (wid >= nEdges) return;
    int r = row[wid];
    int c = col[wid];
    float4 v = ((const float4*)(xsrc + (size_t)r * HID))[lane];
    float* dst = agg + (size_t)c * HID + lane * 4;
    atomicAddF(dst + 0, v.x);
    atomicAddF(dst + 1, v.y);
    atomicAddF(dst + 2, v.z);
    atomicAddF(dst + 3, v.w);
}

// stage a [rows x 128] row-major matrix into LDS in WMMA-B pair layout:
//   lds[(（r/2)*128 + c)*2 + (r&1)] = W[r][c]
// -> B fragment {W[kk][col], W[kk+1][col]} is one aligned 8-byte LDS word.
__device__ __forceinline__ void stage_pairs(const float* __restrict__ W,
                                            float* __restrict__ ldsW,
                                            int nElems, int tid, int nthr) {
    for (int i = tid; i < nElems; i += nthr) {
        int r = i >> 7;
        int c = i & 127;
        ldsW[(((r >> 1) * HID) + c) * 2 + (r & 1)] = W[i];
    }
}

// ---------------- per-layer node transform (WMMA fp32) ---------------------
// x_out = relu( (sagg/deg + (eagg/deg)@Wes + bes) @ Ws + bs ), in-place OK.
// Block = 128 threads = 4 waves; each wave owns TWO 16-node tiles (32 nodes).
__global__ __launch_bounds__(128) void gp_layer_kernel(
    const float* __restrict__ sagg,   // [N,128] scatter-aggregated features
    const float* __restrict__ eagg,   // [N,16]
    const float* __restrict__ deg,    // [N]
    const float* __restrict__ Wes,    // [16,128]
    const float* __restrict__ bes,    // [128]
    const float* __restrict__ Wsl,    // [128,128]
    const float* __restrict__ bsl,    // [128]
    float* __restrict__ xout,         // [N,128], may alias sagg
    int nNodes) {
    extern __shared__ float lds[];
    float* ldsW   = lds;                    // 128*128 (pair layout)
    float* ldsWes = ldsW + HID * HID;       // 16*128  (pair layout)
    float* ldsBs  = ldsWes + EDIM * HID;    // 128
    float* ldsBes = ldsBs + HID;            // 128
    float* ldsMin = ldsBes + HID;           // 4 waves * 2 tiles * 16*128

    int tid = threadIdx.x;
    stage_pairs(Wsl, ldsW, HID * HID, tid, 128);
    stage_pairs(Wes, ldsWes, EDIM * HID, tid, 128);
    if (tid < HID) { ldsBs[tid] = bsl[tid]; ldsBes[tid] = bes[tid]; }
    __syncthreads();

    int wave = tid >> 5;
    int lane = tid & 31;
    int laneN = lane & 15;
    int hi = lane >> 4;                       // 0: lanes 0-15, 1: lanes 16-31
    int job = blockIdx.x * 4 + wave;          // 32 nodes per job
    bool active = (job * 32) < nNodes;
    float* myMin = ldsMin + wave * (2 * 16 * HID);

    if (active) {
        // ---- stage 1: Min = (eagg*invd) @ Wes + (sagg*invd + bes) ----
        #pragma unroll
        for (int mb = 0; mb < 2; ++mb) {
            int node0 = job * 32 + mb * 16;
            float invdA = 1.0f / fmaxf(deg[node0 + laneN], 1.0f);
            float invdC[8];
            #pragma unroll
            for (int r = 0; r < 8; ++r)
                invdC[r] = 1.0f / fmaxf(deg[node0 + r + hi * 8], 1.0f);

            const float* erow = eagg + (size_t)(node0 + laneN) * EDIM;
            float* mTile = myMin + mb * (16 * HID);
            #pragma unroll
            for (int nt = 0; nt < 8; ++nt) {
                int colC = nt * 16 + laneN;
                v8f acc;
                #pragma unroll
                for (int r = 0; r < 8; ++r) {
                    int m = r + hi * 8;
                    acc[r] = sagg[(size_t)(node0 + m) * HID + colC] * invdC[r]
                           + ldsBes[colC];
                }
                #pragma unroll
                for (int k = 0; k < EDIM / 4; ++k) {   // K = 16 -> 4 steps
                    int kk = k * 4 + hi * 2;           // kpair = kk/2
                    v2f a = *(const v2f*)&erow[kk];
                    a.x *= invdA;
                    a.y *= invdA;
                    v2f b = *(const v2f*)&ldsWes[((kk >> 1) * HID + colC) * 2];
                    acc = wmma_f32_4(a, b, acc);
                }
                #pragma unroll
                for (int r = 0; r < 8; ++r)
                    mTile[(r + hi * 8) * HID + colC] = acc[r];
            }
        }
    }
    __syncthreads();

    if (active) {
        // ---- stage 2: x = relu(Min @ Ws + bs), K = 128 -> 32 steps,
        //      2 M-tiles per wave -> each B fragment feeds 2 WMMAs ----
        v8f acc[2][8];
        #pragma unroll
        for (int nt = 0; nt < 8; ++nt) {
            float bv = ldsBs[nt * 16 + laneN];
            #pragma unroll
            for (int r = 0; r < 8; ++r) { acc[0][nt][r] = bv; acc[1][nt][r] = bv; }
        }
        #pragma unroll 4
        for (int k = 0; k < HID / 4; ++k) {
            int kk = k * 4 + hi * 2;
            v2f a0 = *(const v2f*)&myMin[laneN * HID + kk];
            v2f a1 = *(const v2f*)&myMin[(16 + laneN) * HID + kk];
            #pragma unroll
            for (int nt = 0; nt < 8; ++nt) {
                int colC = nt * 16 + laneN;
                v2f b = *(const v2f*)&ldsW[((kk >> 1) * HID + colC) * 2];
                acc[0][nt] = wmma_f32_4(a0, b, acc[0][nt]);
                acc[1][nt] = wmma_f32_4(a1, b, acc[1][nt]);
            }
        }
        #pragma unroll
        for (int mb = 0; mb < 2; ++mb) {
            int node0 = job * 32 + mb * 16;
            #pragma unroll
            for (int nt = 0; nt < 8; ++nt) {
                int colC = nt * 16 + laneN;
                #pragma unroll
                for (int r = 0; r < 8; ++r) {
                    int m = r + hi * 8;
                    xout[(size_t)(node0 + m) * HID + colC] =
                        fmaxf(acc[mb][nt][r], 0.0f);
                }
            }
        }
    }
}

// ------------- output projection (WMMA) + fused graph pooling --------------
__global__ __launch_bounds__(128) void gp_out_kernel(
    const float* __restrict__ x,      // [N,128]
    const float* __restrict__ Wo,     // [128,128]
    const float* __restrict__ bo,     // [128]
    const int* __restrict__ batch,    // [N]
    float* __restrict__ outN,         // [N,128]
    float* __restrict__ gsum,         // [64,128]
    int nNodes) {
    extern __shared__ float lds[];
    float* ldsW  = lds;               // 128*128 (pair layout)
    float* ldsBo = ldsW + HID * HID;  // 128

    int tid = threadIdx.x;
    stage_pairs(Wo, ldsW, HID * HID, tid, 128);
    if (tid < HID) ldsBo[tid] = bo[tid];
    __syncthreads();

    int wave = tid >> 5;
    int lane = tid & 31;
    int laneN = lane & 15;
    int hi = lane >> 4;
    int job = blockIdx.x * 4 + wave;
    if (job * 32 >= nNodes) return;
    int nodeA = job * 32;
    int nodeB = nodeA + 16;

    v8f acc[2][8];
    #pragma unroll
    for (int nt = 0; nt < 8; ++nt) {
        float bv = ldsBo[nt * 16 + laneN];
        #pragma unroll
        for (int r = 0; r < 8; ++r) { acc[0][nt][r] = bv; acc[1][nt][r] = bv; }
    }
    const float* arow0 = x + (size_t)(nodeA + laneN) * HID;
    const float* arow1 = x + (size_t)(nodeB + laneN) * HID;
    #pragma unroll 4
    for (int k = 0; k < HID / 4; ++k) {
        int kk = k * 4 + hi * 2;
        v2f a0 = *(const v2f*)&arow0[kk];
        v2f a1 = *(const v2f*)&arow1[kk];
        #pragma unroll
        for (int nt = 0; nt < 8; ++nt) {
            int colC = nt * 16 + laneN;
            v2f b = *(const v2f*)&ldsW[((kk >> 1) * HID + colC) * 2];
            acc[0][nt] = wmma_f32_4(a0, b, acc[0][nt]);
            acc[1][nt] = wmma_f32_4(a1, b, acc[1][nt]);
        }
    }
    #pragma unroll
    for (int mb = 0; mb < 2; ++mb) {
        int node0 = (mb == 0) ? nodeA : nodeB;
        #pragma unroll
        for (int r = 0; r < 8; ++r) {
            int m = r + hi * 8;
            int g = batch[node0 + m];
            #pragma unroll
            for (int nt = 0; nt < 8; ++nt) {
                int colC = nt * 16 + laneN;
                float v = acc[mb][nt][r];
                outN[(size_t)(node0 + m) * HID + colC] = v;
                atomicAddF(&gsum[(size_t)g * HID + colC], v);
            }
        }
    }
}

__global__ __launch_bounds__(256) void gp_pool_finalize_kernel(
    const float* __restrict__ gsum, const float* __restrict__ cnt,
    float* __restrict__ outG) {
    int i = blockIdx.x * blockDim.x + threadIdx.x;
    if (i < NGRAPH * HID) outG[i] = gsum[i] / cnt[i >> 7];
}

// ---------------------------------------------------------------------------
extern "C" void kernel_launch(void* const* d_in, const int* in_sizes, int n_in,
                              void* d_out, int out_size, void* d_ws, size_t ws_size,
                              hipStream_t stream) {
    const int N = NNODES, E = NEDGES, G = NGRAPH;
    const float* x     = (const float*)d_in[0];
    const int*   eidx  = (const int*)d_in[1];       // row = eidx, col = eidx+E
    const float* eattr = (const float*)d_in[2];
    const int*   batch = (const int*)d_in[3];
    const float* Ws    = (const float*)d_in[4];     // [3][128][128]
    const float* bs    = (const float*)d_in[5];     // [3][128]
    const float* Wes   = (const float*)d_in[6];     // [3][16][128]
    const float* bes   = (const float*)d_in[7];     // [3][128]
    const float* Wo    = (const float*)d_in[8];
    const float* bo    = (const float*)d_in[9];
    const int* row = eidx;
    const int* col = eidx + E;

    float* outNodes = (float*)d_out;
    float* outGraph = outNodes + (size_t)N * HID;

    // workspace layout
    float* aggA = (float*)d_ws;                         // N*128
    float* aggB = aggA + (size_t)N * HID;               // N*128
    float* eagg = aggB + (size_t)N * HID;               // N*16
    float* deg  = eagg + (size_t)N * EDIM;              // N
    float* gsum = deg + N;                              // 64*128
    float* cnt  = gsum + (size_t)G * HID;               // 64

    const int Z = 256;
    // zero: deg, eagg, aggA, gsum+cnt (contiguous tail)
    gp_zero_kernel<<<2048, Z, 0, stream>>>(deg, (long)N);
    gp_zero_kernel<<<2048, Z, 0, stream>>>(eagg, (long)N * EDIM);
    gp_zero_kernel<<<2048, Z, 0, stream>>>(aggA, (long)N * HID);
    gp_zero_kernel<<<64, Z, 0, stream>>>(gsum, (long)(G * HID + G));

    // one-time edge-feature aggregation + degrees + graph counts
    gp_edge_pre_kernel<<<(E * 16 + Z - 1) / Z, Z, 0, stream>>>(eattr, col, eagg, deg, E);
    gp_count_kernel<<<(N + Z - 1) / Z, Z, 0, stream>>>(batch, cnt, N);

    const int jobs = N / 32;                         // 3125 (N % 32 == 0)
    const int lblocks = (jobs + 3) / 4;              // 4 waves per 128-thr block
    const size_t ldsLayer =
        (size_t)(HID * HID + EDIM * HID + HID + HID + 4 * 2 * 16 * HID) * sizeof(float);
    const size_t ldsOut = (size_t)(HID * HID + HID) * sizeof(float);
    const int sblocks = (E + 7) / 8;                 // 8 edges per 256-thr block

    // layer 0: scatter x -> aggA, transform in place
    gp_scatter_kernel<<<sblocks, Z, 0, stream>>>(x, aggA, row, col, E);
    gp_layer_kernel<<<lblocks, 128, ldsLayer, stream>>>(
        aggA, eagg, deg, Wes + 0 * EDIM * HID, bes + 0 * HID,
        Ws + 0 * HID * HID, bs + 0 * HID, aggA, N);

    // layer 1
    gp_zero_kernel<<<2048, Z, 0, stream>>>(aggB, (long)N * HID);
    gp_scatter_kernel<<<sblocks, Z, 0, stream>>>(aggA, aggB, row, col, E);
    gp_layer_kernel<<<lblocks, 128, ldsLayer, stream>>>(
        aggB, eagg, deg, Wes + 1 * EDIM * HID, bes + 1 * HID,
        Ws + 1 * HID * HID, bs + 1 * HID, aggB, N);

    // layer 2
    gp_zero_kernel<<<2048, Z, 0, stream>>>(aggA, (long)N * HID);
    gp_scatter_kernel<<<sblocks, Z, 0, stream>>>(aggB, aggA, row, col, E);
    gp_layer_kernel<<<lblocks, 128, ldsLayer, stream>>>(
        aggA, eagg, deg, Wes + 2 * EDIM * HID, bes + 2 * HID,
        Ws + 2 * HID * HID, bs + 2 * HID, aggA, N);

    // output projection + graph pooling
    gp_out_kernel<<<lblocks, 128, ldsOut, stream>>>(aggA, Wo, bo, batch, outNodes, gsum, N);
    gp_pool_finalize_kernel<<<(G * HID + Z - 1) / Z, Z, 0, stream>>>(gsum, cnt, outGraph);
}